// GCN_12558484373611
// MI455X (gfx1250) — compile-verified
//
#include <hip/hip_runtime.h>

// ---------------------------------------------------------------------------
// GCN 2-layer forward for MI455X (gfx1250, wave32).
//   Layer: h = relu( scatter( (x@W1)[src] * norm ) + b1 )
//          out =      scatter( (h@W2)[src] * norm ) + b2
// GEMM uses V_WMMA_F32_16X16X4_F32 (full f32 precision on the matrix cores).
// ---------------------------------------------------------------------------

typedef __attribute__((ext_vector_type(2))) float v2f;
typedef __attribute__((ext_vector_type(8))) float v8f;

#define FEAT 512   // F_IN == H == 512 for this model

// ---------------- degree / normalization ----------------------------------

__global__ void deg_init_kernel(float* __restrict__ deg, int n) {
    int i = blockIdx.x * blockDim.x + threadIdx.x;
    if (i < n) deg[i] = 1.0f;                  // self-loop contributes 1
}

__global__ void deg_count_kernel(const int* __restrict__ dst,
                                 float* __restrict__ deg, int e) {
    int i = blockIdx.x * blockDim.x + threadIdx.x;
    if (i < e) atomicAdd(&deg[dst[i]], 1.0f);
}

__global__ void deg_rsqrt_kernel(const float* __restrict__ deg,
                                 float* __restrict__ dinv, int n) {
    int i = blockIdx.x * blockDim.x + threadIdx.x;
    if (i < n) dinv[i] = rsqrtf(deg[i]);       // deg >= 1 always
}

// ---------------- f32 WMMA GEMM:  C[M,512] = A[M,512] @ B[512,512] ---------
// Block = 256 threads = 8 waves. Wave w computes C rows [16*(8*bx+w)) .. +16)
// and a 64-wide column strip at n0 = 64*by. K-loop steps by 4 using
// V_WMMA_F32_16X16X4_F32.
//
// f32 A (16x4) lane layout: lanes 0-15 hold M=lane, K={0,1}; lanes 16-31 hold
// M=lane-16, K={2,3}. B (4x16) mirrors the K split across lane halves. C/D is
// the standard 8-VGPR 16x16 layout (VGPR r: M=r / r+8 across lane halves).

__global__ __launch_bounds__(256)
void gemm_wmma_f32_kernel(const float* __restrict__ A,
                          const float* __restrict__ B,
                          float* __restrict__ C, int M) {
    const int K = FEAT, N = FEAT;
    const int wave = threadIdx.x >> 5;
    const int lane = threadIdx.x & 31;

    const int mTile = blockIdx.x * 8 + wave;        // 16-row tile index
    if (mTile * 16 >= M) return;                    // wave-uniform guard
    const int n0 = blockIdx.y * 64;

    const int am = lane & 15;                       // A row inside tile
    const int kh = (lane >> 4) * 2;                 // K sub-offset: 0 or 2
    const int bn = lane & 15;                       // B column inside tile

    const float* __restrict__ Arow = A + (size_t)(mTile * 16 + am) * K + kh;

    v8f acc0 = {}, acc1 = {}, acc2 = {}, acc3 = {};

    #pragma unroll 4
    for (int k = 0; k < K; k += 4) {
        v2f a = *(const v2f*)(Arow + k);            // A[m][k+kh], A[m][k+kh+1]

        const float* __restrict__ Bp = B + (size_t)(k + kh) * N + n0 + bn;
        v2f b0, b1, b2, b3;
        b0.x = Bp[0];       b0.y = Bp[N];
        b1.x = Bp[16];      b1.y = Bp[N + 16];
        b2.x = Bp[32];      b2.y = Bp[N + 32];
        b3.x = Bp[48];      b3.y = Bp[N + 48];

        acc0 = __builtin_amdgcn_wmma_f32_16x16x4_f32(false, a, false, b0,
                                                     (short)0, acc0, false, false);
        acc1 = __builtin_amdgcn_wmma_f32_16x16x4_f32(false, a, false, b1,
                                                     (short)0, acc1, false, false);
        acc2 = __builtin_amdgcn_wmma_f32_16x16x4_f32(false, a, false, b2,
                                                     (short)0, acc2, false, false);
        acc3 = __builtin_amdgcn_wmma_f32_16x16x4_f32(false, a, false, b3,
                                                     (short)0, acc3, false, false);
    }

    // C/D layout: VGPR r, lanes 0-15 -> row r, lanes 16-31 -> row r+8.
    const int rbase = mTile * 16 + (lane >> 4) * 8;
    const int ccol  = n0 + (lane & 15);
    #pragma unroll
    for (int r = 0; r < 8; ++r) {
        float* __restrict__ Cp = C + (size_t)(rbase + r) * N + ccol;
        Cp[0]  = acc0[r];
        Cp[16] = acc1[r];
        Cp[32] = acc2[r];
        Cp[48] = acc3[r];
    }
}

// ---------------- self-loop init:  out = h * dinv[i]^2 (+ bias) ------------

__global__ void self_init_kernel(const float* __restrict__ h,
                                 const float* __restrict__ dinv,
                                 const float* __restrict__ bias,
                                 float* __restrict__ out, int n) {
    long long idx = (long long)blockIdx.x * blockDim.x + threadIdx.x;
    if (idx >= (long long)n * FEAT) return;
    int node = (int)(idx >> 9);
    int f    = (int)(idx & (FEAT - 1));
    float w = dinv[node];
    w = w * w;
    float b = bias ? bias[f] : 0.0f;
    out[idx] = h[idx] * w + b;
}

// ---------------- edge scatter:  out[dst] += h[src] * norm -----------------
// 128 threads per edge, each handling one float4 chunk of the 512-feature row.

__global__ void scatter_kernel(const float* __restrict__ h,
                               const int* __restrict__ src,
                               const int* __restrict__ dst,
                               const float* __restrict__ dinv,
                               float* __restrict__ out, int e) {
    long long idx = (long long)blockIdx.x * blockDim.x + threadIdx.x;
    long long edge = idx >> 7;                      // FEAT/4 = 128 chunks/edge
    int chunk = (int)(idx & 127);
    if (edge >= e) return;

    int s = src[edge];
    int d = dst[edge];
    float norm = dinv[s] * dinv[d];

    const float4* __restrict__ hp =
        (const float4*)(h + (size_t)s * FEAT) + chunk;
    float4 v = *hp;

    float* __restrict__ op = out + (size_t)d * FEAT + chunk * 4;
    atomicAdd(op + 0, v.x * norm);
    atomicAdd(op + 1, v.y * norm);
    atomicAdd(op + 2, v.z * norm);
    atomicAdd(op + 3, v.w * norm);
}

// ---------------- bias + relu ----------------------------------------------

__global__ void bias_relu_kernel(const float* __restrict__ agg,
                                 const float* __restrict__ bias,
                                 float* __restrict__ out, int n) {
    long long idx = (long long)blockIdx.x * blockDim.x + threadIdx.x;
    if (idx >= (long long)n * FEAT) return;
    float v = agg[idx] + bias[idx & (FEAT - 1)];
    out[idx] = v > 0.0f ? v : 0.0f;
}

// ---------------------------------------------------------------------------

extern "C" void kernel_launch(void* const* d_in, const int* in_sizes, int n_in,
                              void* d_out, int out_size, void* d_ws, size_t ws_size,
                              hipStream_t stream) {
    const float* x  = (const float*)d_in[0];
    const int*   ei = (const int*)d_in[1];      // [2, E] edge index
    const float* W1 = (const float*)d_in[2];
    const float* b1 = (const float*)d_in[3];
    const float* W2 = (const float*)d_in[4];
    const float* b2 = (const float*)d_in[5];
    float* out = (float*)d_out;

    const int N = in_sizes[0] / FEAT;           // 100000 nodes
    const int E = in_sizes[1] / 2;              // 800000 edges
    const int* src = ei;
    const int* dst = ei + E;

    // workspace layout
    float* deg  = (float*)d_ws;                 // N
    float* dinv = deg + N;                      // N
    float* bufA = dinv + N;                     // N * FEAT
    float* bufB = bufA + (size_t)N * FEAT;      // N * FEAT

    const int T = 256;
    const long long nodeFeat = (long long)N * FEAT;
    const unsigned nfBlocks  = (unsigned)((nodeFeat + T - 1) / T);
    const unsigned scBlocks  = (unsigned)(((long long)E * (FEAT / 4) + T - 1) / T);
    const int mTiles = (N + 15) / 16;
    dim3 gemmGrid((mTiles + 7) / 8, FEAT / 64);

    // --- normalization -----------------------------------------------------
    deg_init_kernel<<<(N + T - 1) / T, T, 0, stream>>>(deg, N);
    deg_count_kernel<<<(E + T - 1) / T, T, 0, stream>>>(dst, deg, E);
    deg_rsqrt_kernel<<<(N + T - 1) / T, T, 0, stream>>>(deg, dinv, N);

    // --- layer 1 -----------------------------------------------------------
    gemm_wmma_f32_kernel<<<gemmGrid, T, 0, stream>>>(x, W1, bufA, N);
    self_init_kernel<<<nfBlocks, T, 0, stream>>>(bufA, dinv, nullptr, bufB, N);
    scatter_kernel<<<scBlocks, T, 0, stream>>>(bufA, src, dst, dinv, bufB, E);
    bias_relu_kernel<<<nfBlocks, T, 0, stream>>>(bufB, b1, bufA, N);

    // --- layer 2 -----------------------------------------------------------
    gemm_wmma_f32_kernel<<<gemmGrid, T, 0, stream>>>(bufA, W2, bufB, N);
    self_init_kernel<<<nfBlocks, T, 0, stream>>>(bufB, dinv, b2, out, N);
    scatter_kernel<<<scBlocks, T, 0, stream>>>(bufB, src, dst, dinv, out, E);
}